// ResidualGNNLayer_46978352284507
// MI455X (gfx1250) — compile-verified
//
#include <hip/hip_runtime.h>

typedef float v2f __attribute__((ext_vector_type(2)));
typedef float v8f __attribute__((ext_vector_type(8)));

#define DFEAT 128
#define BN_EPS 1e-5f

// ---------------- degree / normalization ----------------
__global__ void k_deg_init(float* __restrict__ deg, int n) {
  int i = blockIdx.x * blockDim.x + threadIdx.x;
  if (i < n) deg[i] = 1.0f;  // self-loop contributes 1
}

__global__ void k_deg_count(const int* __restrict__ dst, float* __restrict__ deg, int e) {
  int i = blockIdx.x * blockDim.x + threadIdx.x;
  if (i < e) atomicAdd(&deg[dst[i]], 1.0f);
}

__global__ void k_dinv_zero_stats(float* __restrict__ deg, float* __restrict__ stats, int n) {
  int i = blockIdx.x * blockDim.x + threadIdx.x;
  if (i < n) deg[i] = rsqrtf(deg[i]);   // in-place deg -> d^{-1/2}
  if (i < 256) stats[i] = 0.0f;         // sum[128] + sumsq[128]
}

// ---------------- xw = x @ W via fp32 WMMA ----------------
// One 16x16 output tile per wave; block = 8 waves covering 16 rows x 128 cols.
__global__ __launch_bounds__(256)
void k_gemm_wmma(const float* __restrict__ x, const float* __restrict__ W,
                 float* __restrict__ xw, int n) {
  const int wave    = threadIdx.x >> 5;
  const int lane    = threadIdx.x & 31;
  const int hi      = lane >> 4;     // lanes 16-31 hold K=2,3 halves
  const int l15     = lane & 15;
  const int rowbase = blockIdx.x * 16;
  const int colbase = wave * 16;

  int arow = rowbase + l15;
  if (arow >= n) arow = n - 1;                 // clamp (keeps EXEC all-ones)
  const float* __restrict__ ap = x + (size_t)arow * DFEAT;
  const float* __restrict__ bp = W + colbase + l15;

  v8f acc = {};
#pragma unroll
  for (int k = 0; k < DFEAT; k += 4) {
    const int k0 = k + (hi ? 2 : 0);
    const int k1 = k + (hi ? 3 : 1);
    v2f a, b;
    a.x = ap[k0];
    a.y = ap[k1];
    b.x = bp[(size_t)k0 * DFEAT];
    b.y = bp[(size_t)k1 * DFEAT];
    acc = __builtin_amdgcn_wmma_f32_16x16x4_f32(
        /*neg_a=*/false, a, /*neg_b=*/false, b,
        /*c_mod=*/(short)0, acc, /*reuse_a=*/false, /*reuse_b=*/false);
  }

#pragma unroll
  for (int g = 0; g < 8; ++g) {
    const int r = rowbase + g + (hi ? 8 : 0);
    if (r < n) xw[(size_t)r * DFEAT + colbase + l15] = acc[g];
  }
}

// ---------------- agg init: bias + self-loop message (into d_out) ----------------
__global__ void k_agg_init(const float* __restrict__ xw, const float* __restrict__ dinv,
                           const float* __restrict__ bias, float* __restrict__ out, int n) {
  long i = (long)blockIdx.x * blockDim.x + threadIdx.x;
  long total = (long)n * DFEAT;
  if (i < total) {
    int row = (int)(i >> 7);
    int col = (int)(i & (DFEAT - 1));
    float di = dinv[row];
    out[i] = bias[col] + xw[i] * di * di;
  }
}

// ---------------- edge scatter: wave per edge, 4 coalesced atomics per lane ----------------
__global__ __launch_bounds__(256)
void k_scatter(const int* __restrict__ src, const int* __restrict__ dst,
               const float* __restrict__ xw, const float* __restrict__ dinv,
               float* __restrict__ out, int e) {
  long t = (long)blockIdx.x * blockDim.x + threadIdx.x;
  int edge = (int)(t >> 5);
  int lane = (int)(t & 31);
  if (edge >= e) return;
  const int s = src[edge];
  const int d = dst[edge];
  const float en = dinv[s] * dinv[d];
  const float* __restrict__ xs = xw + (size_t)s * DFEAT;
  float* __restrict__ od = out + (size_t)d * DFEAT;
#pragma unroll
  for (int j = 0; j < 4; ++j) {
    const int c = j * 32 + lane;   // 32 contiguous lanes per group -> coalesced
    atomicAdd(&od[c], xs[c] * en);
  }
}

// ---------------- BN column statistics (partial sums + atomics) ----------------
__global__ __launch_bounds__(DFEAT)
void k_bn_partial(const float* __restrict__ out, float* __restrict__ sum,
                  float* __restrict__ sumsq, int n) {
  const int col = threadIdx.x;   // 128 threads == 128 columns -> coalesced row loads
  float s = 0.0f, s2 = 0.0f;
  for (int r = blockIdx.x; r < n; r += gridDim.x) {
    const float v = out[(size_t)r * DFEAT + col];
    s  += v;
    s2 += v * v;
  }
  atomicAdd(&sum[col], s);
  atomicAdd(&sumsq[col], s2);
}

// ---------------- finalize: BN affine + PReLU + residual (in-place on d_out) ----------------
__global__ void k_final(float* __restrict__ out, const float* __restrict__ x,
                        const float* __restrict__ sum, const float* __restrict__ sumsq,
                        const float* __restrict__ gamma, const float* __restrict__ beta,
                        const float* __restrict__ aprelu, int n) {
  long i = (long)blockIdx.x * blockDim.x + threadIdx.x;
  long total = (long)n * DFEAT;
  if (i < total) {
    const int col = (int)(i & (DFEAT - 1));
    const float invN = 1.0f / (float)n;
    const float mean = sum[col] * invN;
    const float var  = sumsq[col] * invN - mean * mean;
    float v = (out[i] - mean) * rsqrtf(var + BN_EPS) * gamma[col] + beta[col];
    const float a = aprelu[0];
    v = (v > 0.0f) ? v : a * v;
    out[i] = v + x[i];
  }
}

extern "C" void kernel_launch(void* const* d_in, const int* in_sizes, int n_in,
                              void* d_out, int out_size, void* d_ws, size_t ws_size,
                              hipStream_t stream) {
  const float* x      = (const float*)d_in[0];
  const float* W      = (const float*)d_in[1];
  const float* bias   = (const float*)d_in[2];
  const float* gamma  = (const float*)d_in[3];
  const float* beta   = (const float*)d_in[4];
  const float* aprelu = (const float*)d_in[5];
  const int*   ei     = (const int*)d_in[6];

  const int n = in_sizes[0] / DFEAT;
  const int e = in_sizes[6] / 2;
  const int* src = ei;
  const int* dst = ei + e;
  float* out = (float*)d_out;

  // workspace layout: deg/dinv [n] | stats [256] | xw [n*DFEAT]
  char* ws = (char*)d_ws;
  float* deg   = (float*)ws;
  size_t off   = (((size_t)n * sizeof(float)) + 255) & ~(size_t)255;
  float* stats = (float*)(ws + off);
  float* sum   = stats;
  float* sumsq = stats + DFEAT;
  float* xw    = stats + 256;

  const long total = (long)n * DFEAT;
  const int tb = 256;

  k_deg_init<<<(n + tb - 1) / tb, tb, 0, stream>>>(deg, n);
  k_deg_count<<<(e + tb - 1) / tb, tb, 0, stream>>>(dst, deg, e);
  k_dinv_zero_stats<<<(n + tb - 1) / tb, tb, 0, stream>>>(deg, stats, n);

  k_gemm_wmma<<<(n + 15) / 16, 256, 0, stream>>>(x, W, xw, n);

  k_agg_init<<<(int)((total + tb - 1) / tb), tb, 0, stream>>>(xw, deg, bias, out, n);
  k_scatter<<<(int)(((long)e * 32 + tb - 1) / tb), tb, 0, stream>>>(src, dst, xw, deg, out, e);

  k_bn_partial<<<1024, DFEAT, 0, stream>>>(out, sum, sumsq, n);
  k_final<<<(int)((total + tb - 1) / tb), tb, 0, stream>>>(out, x, sum, sumsq, gamma, beta, aprelu, n);
}